// MPNLWithLogitsLoss_36447092474535
// MI455X (gfx1250) — compile-verified
//
#include <hip/hip_runtime.h>
#include <stdint.h>

// ---------------------------------------------------------------------------
// MPNL-with-logits loss, MI455X (gfx1250) version.
//
// per_row = count_pos * logsumexp(row) - sum_{label!=0} logits[row, j]
// out     = mean(per_row)
//
// Memory-bound (AI ~ 0.4 FLOP/B): optimize the data path, not math.
// CDNA5 features used:
//   * GLOBAL_LOAD_ASYNC_TO_LDS_B128 (ASYNCcnt-tracked DMA of the logits row
//     into LDS, overlapped with the int64 label stream)
//   * s_wait_asynccnt
//   * non-temporal 128-bit label loads (data never re-read; bypass L2 rinse)
//   * wave32 shuffle reductions (warpSize == 32 on gfx1250)
// ---------------------------------------------------------------------------

#define B_ROWS 16384
#define P_COLS 4096
#define TPB    256
#define WAVES  (TPB / 32)

typedef unsigned int u32x4 __attribute__((ext_vector_type(4)));

__device__ __forceinline__ float waveReduceAdd(float v) {
#pragma unroll
  for (int m = 16; m > 0; m >>= 1) v += __shfl_xor(v, m, 32);
  return v;
}

__device__ __forceinline__ float waveReduceMax(float v) {
#pragma unroll
  for (int m = 16; m > 0; m >>= 1) v = fmaxf(v, __shfl_xor(v, m, 32));
  return v;
}

__global__ __launch_bounds__(TPB) void mpnl_row_kernel(
    const float* __restrict__ logits,
    const unsigned long long* __restrict__ labels,
    float* __restrict__ row_out) {
  __shared__ float sLog[P_COLS];     // 16 KB staged logits row
  __shared__ float sMax[WAVES];
  __shared__ float sS[WAVES], sP[WAVES], sC[WAVES];

  const int row = blockIdx.x;
  const int tid = threadIdx.x;
  const float* grow = logits + (size_t)row * P_COLS;

  // LDS byte offset of sLog: low 32 bits of the flat (shared-aperture) address
  // are the LDS offset on gfx1250 (aperture in addr[63:32]).
  const uint32_t lds_base = (uint32_t)(uintptr_t)(&sLog[0]);

  // Async DMA: 4 x b128 per thread; instruction k moves bytes
  // [k*4096 + tid*16, +16) of the row. 4 x 4 KB = whole 16 KB row.
#pragma unroll
  for (int k = 0; k < 4; ++k) {
    uint32_t off = (uint32_t)(k * 4096 + tid * 16);    // byte offset in row
    uint32_t lds_addr = lds_base + off;
    asm volatile("global_load_async_to_lds_b128 %0, %1, %2"
                 :
                 : "v"(lds_addr), "v"(off), "s"(grow)
                 : "memory");
  }

  // Overlap: stream labels (int64, streaming -> non-temporal), build masks.
  // uint4 element j covers labels [2j, 2j+1]; j = k*256 + tid.
  const u32x4* lab = (const u32x4*)(labels + (size_t)row * P_COLS);
  uint32_t mask[8];
  float cnt = 0.0f;
#pragma unroll
  for (int k = 0; k < 8; ++k) {
    u32x4 L = __builtin_nontemporal_load(&lab[k * TPB + tid]);
    uint32_t m0 = ((L.x | L.y) != 0u) ? 1u : 0u;
    uint32_t m1 = ((L.z | L.w) != 0u) ? 1u : 0u;
    mask[k] = m0 | (m1 << 1);
    cnt += (float)(m0 + m1);
  }

  // Wait for the tensor/async DMA, then make LDS visible to the block.
  asm volatile("s_wait_asynccnt 0x0" ::: "memory");
  __syncthreads();

  // Pass 1 from LDS into registers: local max and masked logit sum.
  // float2 element j covers floats [2j, 2j+1]; same j mapping as labels.
  const float2* sF2 = (const float2*)sLog;
  float2 v[8];
  float lmax = -__builtin_inff();
  float psum = 0.0f;
#pragma unroll
  for (int k = 0; k < 8; ++k) {
    v[k] = sF2[k * TPB + tid];
    lmax = fmaxf(lmax, fmaxf(v[k].x, v[k].y));
    if (mask[k] & 1u) psum += v[k].x;
    if (mask[k] & 2u) psum += v[k].y;
  }

  // Row max across 8 waves.
  lmax = waveReduceMax(lmax);
  const int wid = tid >> 5, lane = tid & 31;
  if (lane == 0) sMax[wid] = lmax;
  __syncthreads();
  float rowmax = sMax[0];
#pragma unroll
  for (int w = 1; w < WAVES; ++w) rowmax = fmaxf(rowmax, sMax[w]);

  // Pass 2 from registers: sum of exponentials.
  float esum = 0.0f;
#pragma unroll
  for (int k = 0; k < 8; ++k)
    esum += __expf(v[k].x - rowmax) + __expf(v[k].y - rowmax);

  esum = waveReduceAdd(esum);
  psum = waveReduceAdd(psum);
  cnt  = waveReduceAdd(cnt);
  if (lane == 0) { sS[wid] = esum; sP[wid] = psum; sC[wid] = cnt; }
  __syncthreads();

  if (tid == 0) {
    float S = 0.0f, Pm = 0.0f, C = 0.0f;
#pragma unroll
    for (int w = 0; w < WAVES; ++w) { S += sS[w]; Pm += sP[w]; C += sC[w]; }
    const float lse = rowmax + __logf(S);
    row_out[row] = C * lse - Pm;
  }
}

// Deterministic final mean over the 16384 per-row losses (no float atomics).
__global__ __launch_bounds__(TPB) void mpnl_mean_kernel(
    const float* __restrict__ row_out, float* __restrict__ out) {
  __shared__ float sS[WAVES];
  const float4* r4 = (const float4*)row_out;   // 16384/4 = 4096 float4
  float s = 0.0f;
  for (int i = threadIdx.x; i < (B_ROWS / 4); i += TPB) {
    float4 x = r4[i];
    s += (x.x + x.y) + (x.z + x.w);
  }
  s = waveReduceAdd(s);
  if ((threadIdx.x & 31) == 0) sS[threadIdx.x >> 5] = s;
  __syncthreads();
  if (threadIdx.x == 0) {
    float t = 0.0f;
#pragma unroll
    for (int w = 0; w < WAVES; ++w) t += sS[w];
    out[0] = t * (1.0f / (float)B_ROWS);
  }
}

extern "C" void kernel_launch(void* const* d_in, const int* in_sizes, int n_in,
                              void* d_out, int out_size, void* d_ws,
                              size_t ws_size, hipStream_t stream) {
  const float* logits = (const float*)d_in[0];
  const unsigned long long* labels = (const unsigned long long*)d_in[1];
  float* row_ws = (float*)d_ws;               // 16384 floats = 64 KB scratch

  mpnl_row_kernel<<<B_ROWS, TPB, 0, stream>>>(logits, labels, row_ws);
  mpnl_mean_kernel<<<1, TPB, 0, stream>>>(row_ws, (float*)d_out);
}